// Model_54941221651106
// MI455X (gfx1250) — compile-verified
//
#include <hip/hip_runtime.h>
#include <math.h>

// GRPO loss: B=4, L=2048, V=32000, beta=0.1
// Memory-bound: 1.05 GB logits streamed once -> ~45us floor @ 23.3 TB/s.
// Hot loop uses v_exp_f32 (__expf) so VALU/trans stays well under the HBM floor.

typedef __attribute__((ext_vector_type(2))) float v2f;
typedef __attribute__((ext_vector_type(8))) float v8f;

#define B_      4
#define L_      2048
#define V_      32000
#define BETA_   0.1f
#define THREADS 256
#define NVEC    (V_ / 4)              // 8000 float4 per row
#define FULL    (NVEC / THREADS)      // 31 full strided iterations
#define REM     (NVEC - FULL*THREADS) // 64 leftover float4 (handled by tid<64)
#define NWAVES  (THREADS / 32)

__global__ void grpo_zero_ws(float* __restrict__ ws) {
    int i = threadIdx.x;
    if (i < 2 * B_) ws[i] = 0.0f;
}

// online-softmax accumulate of one float4 (fast hardware exp)
__device__ __forceinline__ void online4(float4 v, float& m, float& s) {
    float xm = fmaxf(fmaxf(v.x, v.y), fmaxf(v.z, v.w));
    if (xm > m) { s *= __expf(m - xm); m = xm; }
    s += __expf(v.x - m) + __expf(v.y - m) + __expf(v.z - m) + __expf(v.w - m);
}

__global__ __launch_bounds__(THREADS)
void grpo_row_kernel(const float* __restrict__ logits,
                     const float* __restrict__ ref_logp,
                     const int*   __restrict__ input_ids,
                     const float* __restrict__ advantages,
                     const float* __restrict__ cmask,
                     float* __restrict__ ws)   // ws[0..B) = numerator, ws[B..2B) = denom
{
    const int row  = blockIdx.x;       // 0 .. B*L-1
    const int tid  = threadIdx.x;
    const int lane = tid & 31;
    const int wid  = tid >> 5;

    const float*  rowp = logits + (size_t)row * V_;
    const float4* rp4  = reinterpret_cast<const float4*>(rowp);

    // target logit (gathered once by thread 0)
    float tgt = 0.0f;
    if (tid == 0) tgt = rowp[input_ids[row]];

    // ---- streaming online softmax over V (b128 loads + uniform prefetch) ----
    float m = -INFINITY, s = 0.0f;
    for (int k = 0; k < FULL; ++k) {
        int i = tid + k * THREADS;
        if (k + 8 < FULL)  // uniform scalar branch
            __builtin_prefetch(rp4 + i + 8 * THREADS, 0, 3);  // global_prefetch_b8
        online4(rp4[i], m, s);
    }
    if (tid < REM)
        online4(rp4[FULL * THREADS + tid], m, s);

    __shared__ float smax[NWAVES];
    __shared__ float st[THREADS];

    // ---- block max: wave32 shuffle reduce, then LDS across 8 waves ----
    float wm = m;
#pragma unroll
    for (int off = 16; off >= 1; off >>= 1)
        wm = fmaxf(wm, __shfl_xor(wm, off, 32));
    if (lane == 0) smax[wid] = wm;
    __syncthreads();
    float M = smax[0];
#pragma unroll
    for (int i = 1; i < NWAVES; ++i) M = fmaxf(M, smax[i]);

    // rescale partial sums to common max (precise, once per thread), park in LDS
    st[tid] = s * expf(m - M);
    __syncthreads();

    // ---- final 256-wide sum via V_WMMA_F32_16X16X4_F32 (wave 0 only) ----
    // B = all-ones => D = A x 1 + C accumulates row-sums of A; summing the 16
    // distinct D rows yields the total of all 64 A entries per WMMA. Only the
    // documented A(16x4 f32) and C/D(16x16 f32) layouts matter; B layout is
    // irrelevant since every element is 1.0.
    if (tid < 32) {
        v8f acc = {};
        v2f onesv; onesv.x = 1.0f; onesv.y = 1.0f;
#pragma unroll
        for (int c = 0; c < THREADS / 64; ++c) {
            v2f a;
            a.x = st[c * 64 + tid];        // one half of the 64-value chunk
            a.y = st[c * 64 + 32 + tid];   // the other half
            acc = __builtin_amdgcn_wmma_f32_16x16x4_f32(
                false, a, false, onesv, (short)0, acc, false, false);
        }
        // lanes 0-15 VGPR r = D[r][n] (rows 0..7), lanes 16-31 = rows 8..15
        float r = acc[0] + acc[1] + acc[2] + acc[3] +
                  acc[4] + acc[5] + acc[6] + acc[7];
        r += __shfl_xor(r, 16, 32);        // total sum in every lane

        if (tid == 0) {
            float logZ = M + logf(r);
            float lp   = tgt - logZ;                    // per-token logp
            int   b    = row / L_;
            float diff = ref_logp[row] - lp;
            float kl   = expf(diff) - diff - 1.0f;      // KL estimator
            float cm   = cmask[row];
            // ratio forward value == 1 (stop-gradient to itself)
            float loss = -(advantages[b] - BETA_ * kl) * cm;
            atomicAdd(&ws[b], loss);
            atomicAdd(&ws[B_ + b], cm);
        }
    }
}

__global__ void grpo_finalize(const float* __restrict__ ws,
                              float* __restrict__ out) {
    if (threadIdx.x == 0) {
        float acc = 0.0f;
#pragma unroll
        for (int b = 0; b < B_; ++b)
            acc += ws[b] / ws[B_ + b];
        out[0] = acc * (1.0f / (float)B_);
    }
}

extern "C" void kernel_launch(void* const* d_in, const int* in_sizes, int n_in,
                              void* d_out, int out_size, void* d_ws, size_t ws_size,
                              hipStream_t stream) {
    const float* logits     = (const float*)d_in[0];
    const float* ref_logp   = (const float*)d_in[1];
    const int*   input_ids  = (const int*)d_in[2];
    const float* advantages = (const float*)d_in[3];
    const float* cmask      = (const float*)d_in[4];
    float* ws  = (float*)d_ws;
    float* out = (float*)d_out;

    grpo_zero_ws<<<1, 32, 0, stream>>>(ws);
    grpo_row_kernel<<<B_ * L_, THREADS, 0, stream>>>(
        logits, ref_logp, input_ids, advantages, cmask, ws);
    grpo_finalize<<<1, 32, 0, stream>>>(ws, out);
}